// SISNRLoss_12833362280914
// MI455X (gfx1250) — compile-verified
//
#include <hip/hip_runtime.h>
#include <hip/hip_bf16.h>
#include <math.h>

typedef __attribute__((ext_vector_type(2))) float v2f;
typedef __attribute__((ext_vector_type(8))) float v8f;

#define BROWS   256
#define TLEN    48000
#define CHUNKS  16
#define CHLEN   (TLEN / CHUNKS)   // 3000 floats per (row, chunk)
#define THREADS 256

// ---------------------------------------------------------------------------
// Pass 1: bandwidth-bound streaming reduction.
// grid = (256 rows, 16 chunks); each block reduces a contiguous 12 KB slice
// of input and target with float4 (global_load_b128) accesses and writes
// 3 partial sums (dot, ||t||^2, ||x||^2) to a fixed slot in d_ws.
// Deterministic: every slot written exactly once per call, no atomics.
// ---------------------------------------------------------------------------
__global__ __launch_bounds__(THREADS)
void sisnr_partials_kernel(const float* __restrict__ input,
                           const float* __restrict__ target,
                           float* __restrict__ partials) {
  const int row = blockIdx.x;
  const int ch  = blockIdx.y;
  const long base = (long)row * TLEN + (long)ch * CHLEN;
  const float4* __restrict__ in4 = (const float4*)(input + base);
  const float4* __restrict__ tg4 = (const float4*)(target + base);
  const int n4 = CHLEN >> 2;   // 750 float4 per block

  float accd = 0.f, acct = 0.f, acci = 0.f;
  for (int idx = threadIdx.x; idx < n4; idx += THREADS) {
    // speculative prefetch two block-strides ahead (global_prefetch_b8);
    // OOB translation failures are silently dropped per ISA 10.5
    __builtin_prefetch((const void*)(in4 + idx + 2 * THREADS), 0, 0);
    __builtin_prefetch((const void*)(tg4 + idx + 2 * THREADS), 0, 0);
    float4 t = tg4[idx];
    float4 x = in4[idx];
    accd = fmaf(t.x, x.x, accd); accd = fmaf(t.y, x.y, accd);
    accd = fmaf(t.z, x.z, accd); accd = fmaf(t.w, x.w, accd);
    acct = fmaf(t.x, t.x, acct); acct = fmaf(t.y, t.y, acct);
    acct = fmaf(t.z, t.z, acct); acct = fmaf(t.w, t.w, acct);
    acci = fmaf(x.x, x.x, acci); acci = fmaf(x.y, x.y, acci);
    acci = fmaf(x.z, x.z, acci); acci = fmaf(x.w, x.w, acci);
  }

  // wave32 butterfly reduction
  for (int off = 16; off > 0; off >>= 1) {
    accd += __shfl_down(accd, off, 32);
    acct += __shfl_down(acct, off, 32);
    acci += __shfl_down(acci, off, 32);
  }

  __shared__ float sd[THREADS / 32], st[THREADS / 32], si[THREADS / 32];
  const int wave = threadIdx.x >> 5;
  if ((threadIdx.x & 31) == 0) { sd[wave] = accd; st[wave] = acct; si[wave] = acci; }
  __syncthreads();
  if (threadIdx.x == 0) {
    float d = 0.f, t = 0.f, i = 0.f;
    for (int w = 0; w < THREADS / 32; ++w) { d += sd[w]; t += st[w]; i += si[w]; }
    const int o = row * CHUNKS + ch;
    partials[0 * BROWS * CHUNKS + o] = d;
    partials[1 * BROWS * CHUNKS + o] = t;
    partials[2 * BROWS * CHUNKS + o] = i;
  }
}

// ---------------------------------------------------------------------------
// Pass 2: one full wave (EXEC all-ones, required for WMMA).
// Each lane folds CHUNK partials for 8 rows into 8 log-ratios, then the 256
// values are summed by the matrix unit: four V_WMMA_F32_16X16X4_F32 ops with
// A = ones(16x4) accumulate column sums of each 4x16 B tile into D. All rows
// of D are identical, and VGPR0 lanes 16-31 replicate row 0, so a full 32-lane
// sum of acc[0] yields 2 * total; loss = -(2*total) / (2*256).
// ---------------------------------------------------------------------------
__global__ __launch_bounds__(32)
void sisnr_final_kernel(const float* __restrict__ partials,
                        float* __restrict__ out) {
  const int lane = threadIdx.x;   // 0..31, full wave
  float lr[8];
#pragma unroll
  for (int j = 0; j < 8; ++j) {
    const int row = lane + 32 * j;
    float d = 0.f, te = 0.f, ie = 0.f;
#pragma unroll
    for (int c = 0; c < CHUNKS; ++c) {
      const int o = row * CHUNKS + c;
      d  += partials[0 * BROWS * CHUNKS + o];
      te += partials[1 * BROWS * CHUNKS + o];
      ie += partials[2 * BROWS * CHUNKS + o];
    }
    const float se = (d * d) / te;   // ||s_target||^2
    const float ne = ie - se;        // ||noise||^2
    lr[j] = 10.0f * log10f(se / ne);
  }

  v2f onesA; onesA[0] = 1.0f; onesA[1] = 1.0f;
  v8f acc = {};
#pragma unroll
  for (int t = 0; t < 4; ++t) {
    v2f b; b[0] = lr[2 * t]; b[1] = lr[2 * t + 1];
    // D = Ones(16x4) x B(4x16) + C  -> adds sum over K of each column
    acc = __builtin_amdgcn_wmma_f32_16x16x4_f32(
        /*neg_a=*/false, onesA, /*neg_b=*/false, b,
        /*c_mod=*/(short)0, acc, /*reuse_a=*/false, /*reuse_b=*/false);
  }

  float v = acc[0];                       // row 0 (lanes 0-15) + row 8 copy (lanes 16-31)
  for (int off = 16; off > 0; off >>= 1) v += __shfl_xor(v, off, 32);
  if (lane == 0) out[0] = -v * (1.0f / 512.0f);  // /2 double count, /256 mean, negate
}

// ---------------------------------------------------------------------------
extern "C" void kernel_launch(void* const* d_in, const int* in_sizes, int n_in,
                              void* d_out, int out_size, void* d_ws, size_t ws_size,
                              hipStream_t stream) {
  (void)in_sizes; (void)n_in; (void)out_size; (void)ws_size;
  const float* input  = (const float*)d_in[0];   // [B, T] f32
  const float* target = (const float*)d_in[1];   // [B, T] f32
  float* out      = (float*)d_out;               // scalar f32
  float* partials = (float*)d_ws;                // 3 * 256 * 16 floats = 48 KB

  dim3 grid(BROWS, CHUNKS);
  sisnr_partials_kernel<<<grid, THREADS, 0, stream>>>(input, target, partials);
  sisnr_final_kernel<<<1, 32, 0, stream>>>(partials, out);
}